// QuantumLayer_27376121544751
// MI455X (gfx1250) — compile-verified
//
#include <hip/hip_runtime.h>
#include <hip/hip_bf16.h>
#include <math.h>

#define N_QUBITS 14
#define N_LAYERS 2
#define N_FEATURES 128
#define BATCH 128
#define DIM (1 << N_QUBITS)   // 16384
#define CIRC_THREADS 512

typedef float v2f __attribute__((ext_vector_type(2)));
typedef float v8f __attribute__((ext_vector_type(8)));

// ---------------------------------------------------------------------------
// Kernel 1: full statevector evolution, one sample per workgroup.
// State stored INTERLEAVED as float2 (re,im) in dynamic LDS (128 KB) so every
// amplitude access is a single ds_load_b64/ds_store_b64 (half the DS issue
// rate vs split re/im). Writes Z expectations (padded to 16) into d_ws.
// ---------------------------------------------------------------------------
__global__ __launch_bounds__(CIRC_THREADS)
void qsim_circuit_kernel(const float* __restrict__ x,        // [BATCH][N_QUBITS]
                         const float* __restrict__ w,        // [N_LAYERS][N_QUBITS][3]
                         float* __restrict__ zout)           // [BATCH][16]
{
    extern __shared__ float smem[];
    float2* st   = (float2*)smem;          // DIM complex amplitudes (128 KB)
    float*  zred = smem + 2 * DIM;         // 16 floats for Z reduction

    const int b   = blockIdx.x;
    const int tid = threadIdx.x;

    // init |0...0>
    for (int i = tid; i < DIM; i += CIRC_THREADS) st[i] = make_float2(0.0f, 0.0f);
    if (tid == 0) st[0] = make_float2(1.0f, 0.0f);
    if (tid < 16) zred[tid] = 0.0f;
    __syncthreads();

    // ---- angle encoding: RY(x[b][q]) on wire q (wire 0 = MSB) ----
    for (int q = 0; q < N_QUBITS; ++q) {
        const int   p      = N_QUBITS - 1 - q;     // bit position from LSB
        const int   stride = 1 << p;
        const float t  = x[b * N_QUBITS + q] * 0.5f;
        const float cc = cosf(t), ss = sinf(t);
        for (int j = tid; j < DIM / 2; j += CIRC_THREADS) {
            const int i0 = ((j >> p) << (p + 1)) | (j & (stride - 1));
            const int i1 = i0 | stride;
            const float2 a0 = st[i0];
            const float2 a1 = st[i1];
            st[i0] = make_float2(cc * a0.x - ss * a1.x, cc * a0.y - ss * a1.y);
            st[i1] = make_float2(ss * a0.x + cc * a1.x, ss * a0.y + cc * a1.y);
        }
        __syncthreads();
    }

    // ---- variational layers: fused U = RZ(w2) * RY(w1) * RX(w0), then CNOT ring ----
    for (int layer = 0; layer < N_LAYERS; ++layer) {
        for (int q = 0; q < N_QUBITS; ++q) {
            const int base = (layer * N_QUBITS + q) * 3;
            const float tx = w[base + 0] * 0.5f;
            const float ty = w[base + 1] * 0.5f;
            const float tz = w[base + 2] * 0.5f;
            const float cx = cosf(tx), sx = sinf(tx);
            const float cy = cosf(ty), sy = sinf(ty);
            const float cz = cosf(tz), sz = sinf(tz);
            // M = RY*RX:
            // M00 = cy*cx + i sy*sx     M01 = -sy*cx - i cy*sx
            // M10 = sy*cx - i cy*sx     M11 =  cy*cx - i sy*sx
            const float m00r =  cy * cx, m00i =  sy * sx;
            const float m01r = -sy * cx, m01i = -cy * sx;
            const float m10r =  sy * cx, m10i = -cy * sx;
            const float m11r =  cy * cx, m11i = -sy * sx;
            // U = RZ*M: row0 *= (cz - i sz), row1 *= (cz + i sz)
            const float u00r = cz * m00r + sz * m00i, u00i = cz * m00i - sz * m00r;
            const float u01r = cz * m01r + sz * m01i, u01i = cz * m01i - sz * m01r;
            const float u10r = cz * m10r - sz * m10i, u10i = cz * m10i + sz * m10r;
            const float u11r = cz * m11r - sz * m11i, u11i = cz * m11i + sz * m11r;

            const int p = N_QUBITS - 1 - q, stride = 1 << p;
            for (int j = tid; j < DIM / 2; j += CIRC_THREADS) {
                const int i0 = ((j >> p) << (p + 1)) | (j & (stride - 1));
                const int i1 = i0 | stride;
                const float2 a = st[i0];
                const float2 c = st[i1];
                st[i0] = make_float2(u00r * a.x - u00i * a.y + u01r * c.x - u01i * c.y,
                                     u00r * a.y + u00i * a.x + u01r * c.y + u01i * c.x);
                st[i1] = make_float2(u10r * a.x - u10i * a.y + u11r * c.x - u11i * c.y,
                                     u10r * a.y + u10i * a.x + u11r * c.y + u11i * c.x);
            }
            __syncthreads();
        }
        // CNOT ring: (0,1)...(12,13),(13,0)
        for (int e = 0; e < N_QUBITS; ++e) {
            const int cw = (e < N_QUBITS - 1) ? e : (N_QUBITS - 1);
            const int tw = (e < N_QUBITS - 1) ? (e + 1) : 0;
            const int pc = N_QUBITS - 1 - cw, pt = N_QUBITS - 1 - tw;
            const int plo = pc < pt ? pc : pt;
            const int phi = pc < pt ? pt : pc;
            for (int j = tid; j < DIM / 4; j += CIRC_THREADS) {
                int v = ((j >> plo) << (plo + 1)) | (j & ((1 << plo) - 1));
                v     = ((v >> phi) << (phi + 1)) | (v & ((1 << phi) - 1));
                const int i0 = v | (1 << pc);      // control = 1, target = 0
                const int i1 = i0 | (1 << pt);     // control = 1, target = 1
                const float2 t0 = st[i0];
                const float2 t1 = st[i1];
                st[i0] = t1;
                st[i1] = t0;
            }
            __syncthreads();
        }
    }

    // ---- Pauli-Z expectations per wire ----
    float acc[N_QUBITS];
#pragma unroll
    for (int q = 0; q < N_QUBITS; ++q) acc[q] = 0.0f;
    for (int i = tid; i < DIM; i += CIRC_THREADS) {
        const float2 v = st[i];
        const float pr = v.x * v.x + v.y * v.y;
#pragma unroll
        for (int q = 0; q < N_QUBITS; ++q)
            acc[q] += ((i >> (N_QUBITS - 1 - q)) & 1) ? -pr : pr;
    }
#pragma unroll
    for (int q = 0; q < N_QUBITS; ++q) atomicAdd(&zred[q], acc[q]);
    __syncthreads();
    if (tid < 16) zout[b * 16 + tid] = (tid < N_QUBITS) ? zred[tid] : 0.0f;
}

// ---------------------------------------------------------------------------
// Kernel 2: FC layer out = Z @ fc_w^T + fc_b as a 128x16x128 f32 GEMM using
// V_WMMA_F32_16X16X4_F32 (4 chained K-steps of 4). One wave per 16x16 tile.
// A (16x4): lanes 0-15 hold M rows with v0/v1 = K0/K1; lanes 16-31 = K2/K3.
// C/D (16x16): VGPR r = row r (lanes 0-15) / row r+8 (lanes 16-31), N = lane&15.
// ---------------------------------------------------------------------------
__global__ __launch_bounds__(32)
void qsim_fc_wmma_kernel(const float* __restrict__ Z,      // [BATCH][16] padded
                         const float* __restrict__ fc_w,   // [N_FEATURES][N_QUBITS]
                         const float* __restrict__ fc_b,   // [N_FEATURES]
                         float* __restrict__ out)          // [BATCH][N_FEATURES]
{
    const int lane = threadIdx.x & 31;
    const int bm   = blockIdx.x;       // batch tile 0..7
    const int bn   = blockIdx.y;       // feature tile 0..7
    const int half = lane >> 4;        // 0: K pair (0,1), 1: K pair (2,3)
    const int l16  = lane & 15;

    v8f c = {};
#pragma unroll
    for (int s = 0; s < 4; ++s) {
        const int k0 = s * 4 + half * 2;
        // A[M][K] = Z[bm*16 + l16][k0 .. k0+1]   (Z already zero-padded to K=16)
        v2f a;
        a.x = Z[(bm * 16 + l16) * 16 + k0];
        a.y = Z[(bm * 16 + l16) * 16 + k0 + 1];
        // B[K][N] = fc_w[bn*16 + N][K], zero beyond K=13
        const int f = bn * 16 + l16;
        v2f bb;
        bb.x = (k0     < N_QUBITS) ? fc_w[f * N_QUBITS + k0]     : 0.0f;
        bb.y = (k0 + 1 < N_QUBITS) ? fc_w[f * N_QUBITS + k0 + 1] : 0.0f;
        c = __builtin_amdgcn_wmma_f32_16x16x4_f32(
                /*neg_a=*/false, a, /*neg_b=*/false, bb,
                /*c_mod=*/(short)0, c, /*reuse_a=*/false, /*reuse_b=*/false);
    }

    const float bias = fc_b[bn * 16 + l16];
#pragma unroll
    for (int r = 0; r < 8; ++r) {
        const int m = r + half * 8;
        out[(bm * 16 + m) * N_FEATURES + bn * 16 + l16] = c[r] + bias;
    }
}

// ---------------------------------------------------------------------------
extern "C" void kernel_launch(void* const* d_in, const int* in_sizes, int n_in,
                              void* d_out, int out_size, void* d_ws, size_t ws_size,
                              hipStream_t stream) {
    const float* x       = (const float*)d_in[0];   // [128][14]
    const float* weights = (const float*)d_in[1];   // [2][14][3]
    const float* fc_w    = (const float*)d_in[2];   // [128][14]
    const float* fc_b    = (const float*)d_in[3];   // [128]
    float*       out     = (float*)d_out;           // [128][128]
    float*       Zpad    = (float*)d_ws;            // [128][16] scratch

    const size_t smem_bytes = (size_t)(2 * DIM + 16) * sizeof(float); // ~128 KB
    qsim_circuit_kernel<<<BATCH, CIRC_THREADS, smem_bytes, stream>>>(x, weights, Zpad);

    dim3 grid(BATCH / 16, N_FEATURES / 16);   // 8 x 8 tiles, one wave each
    qsim_fc_wmma_kernel<<<grid, 32, 0, stream>>>(Zpad, fc_w, fc_b, out);
}